// BinaryTreeLogicNet_69157563401062
// MI455X (gfx1250) — compile-verified
//
#include <hip/hip_runtime.h>
#include <cstdint>

// ---------------------------------------------------------------------------
// BinaryTreeLogicNet on MI455X (gfx1250)
//   - memory-bound: 256MB / 23.3TB/s ~= 11us floor
//   - no matmul structure (abs/pow/max are elementwise) -> WMMA not applicable
//   - CDNA5 Tensor Data Mover (tensor_load_to_lds) stages each 4KB row into
//     LDS with hardware padding (lane-chunk transpose for free), double
//     buffered per wave, synchronized with s_wait_tensorcnt / s_wait_dscnt
// ---------------------------------------------------------------------------

#define TPB           128      // 4 waves of 32
#define WAVES         4
#define RPW           8        // rows per wave (pipelined)
#define CHUNK_STRIDE  144      // 128B data + 16B TDM pad per lane
#define BUF_BYTES     (32 * CHUNK_STRIDE)   // 4608 B per row buffer
#define N_FEAT        1024
#define EPSV          1e-6f
#define LOG2E         1.4426950408889634f

typedef unsigned int __attribute__((ext_vector_type(4))) v4u;
typedef int          __attribute__((ext_vector_type(8))) v8i;
typedef int          __attribute__((ext_vector_type(4))) v4i;

// v_exp_f32 / v_log_f32 are base-2 on AMDGCN.
__device__ __forceinline__ float fast_exp2(float x) { return __builtin_amdgcn_exp2f(x); }
__device__ __forceinline__ float fast_log2(float x) { return __builtin_amdgcn_logf(x); }

__device__ __forceinline__ float sigmoidf_dev(float x) {
    // 1/(1+e^-x) = rcp(1+2^(-x*log2e)); v_rcp_f32 instead of IEEE divide
    return __builtin_amdgcn_rcpf(1.0f + fast_exp2(-x * LOG2E));
}

// a^lam * b^(1-lam) + (1-lam)*max(a,b), a=|x0*w0|+eps, b=|x1*w1|+eps
// oml = 1-lam precomputed in the const table.
__device__ __forceinline__ float combine(float x0, float x1,
                                         float w0, float w1,
                                         float lam, float oml) {
    float a  = fabsf(x0 * w0) + EPSV;
    float b  = fabsf(x1 * w1) + EPSV;
    float la = fast_log2(a);
    float lb = fast_log2(b);
    float g  = fast_exp2(lb + lam * (la - lb));   // a^lam * b^(1-lam)
    return g + oml * fmaxf(a, b);
}

// One TDM op: DMA a 4KB row (1024 f32) from global into LDS at lds_byte,
// inserting 4 DWORDs of padding after every 32 DWORDs (pad_interval=4 -> 32dw,
// pad_amount=3 -> 4dw). Result: lane L's 32 floats land at lds_byte + L*144,
// 16B aligned for ds_load_b128 with bank-skewed starts.
__device__ __forceinline__ void tdm_load_row(unsigned lds_byte, const float* gsrc) {
    unsigned long long ga = (unsigned long long)(uintptr_t)gsrc;
    v4u g0;
    g0[0] = 1u;                                      // count=1, user descriptor
    g0[1] = lds_byte;                                // lds_addr
    g0[2] = (unsigned)(ga & 0xffffffffull);          // global_addr[31:0]
    g0[3] = (unsigned)((ga >> 32) & 0x01ffffffu)     // global_addr[56:32]
          | (2u << 30);                              // type=2 (image)
    v8i g1;
    g1[0] = (2 << 16)      // data_size = 4B
          | (1 << 20)      // pad_enable
          | (4 << 22)      // pad_interval: 32 DWORDs
          | (3 << 25);     // pad_amount: 4 DWORDs
    g1[1] = (int)((N_FEAT & 0xffffu) << 16);         // tensor_dim0 lo16 @bits[63:48]
    g1[2] = (int)((N_FEAT >> 16) | (1u << 16));      // tensor_dim0 hi16, tensor_dim1=1
    g1[3] = (int)(N_FEAT << 16);                     // tile_dim0=1024 @bits[127:112]
    g1[4] = 1;                                       // tile_dim1=1
    g1[5] = N_FEAT;                                  // tensor_dim0_stride
    g1[6] = 0;
    g1[7] = 0;
    v4i z = {0, 0, 0, 0};
#if defined(__clang_major__) && (__clang_major__ >= 23)
    v8i z8 = {0, 0, 0, 0, 0, 0, 0, 0};
    __builtin_amdgcn_tensor_load_to_lds(g0, g1, z, z, z8, 0);
#else
    __builtin_amdgcn_tensor_load_to_lds(g0, g1, z, z, 0);
#endif
}

__global__ __launch_bounds__(TPB)
void tree_logic_kernel(const float* __restrict__ x,
                       const float* __restrict__ w,      // (1023,2) flat
                       const float* __restrict__ bias,   // (1023,)
                       const float* __restrict__ fcw,
                       const float* __restrict__ fcb,
                       float* __restrict__ out) {
    __shared__ float4 lwq[1024];                               // {w0,w1,lam,1-lam}
    __shared__ __align__(16) char xbuf[WAVES][2][BUF_BYTES];   // double-buffered rows

    const int tid = threadIdx.x;

    // Per-block constant prep: lam = sigmoid(bias), amortized over 32 rows/block.
    for (int p = tid; p < 1023; p += TPB) {
        float w0  = w[2 * p];
        float w1  = w[2 * p + 1];
        float lam = sigmoidf_dev(bias[p]);
        lwq[p] = make_float4(w0, w1, lam, 1.0f - lam);
    }
    if (tid == 0) lwq[1023] = make_float4(0.0f, 0.0f, 0.5f, 0.5f); // clamp slot

    const int  lane = tid & 31;
    const int  wid  = __builtin_amdgcn_readfirstlane(tid >> 5);
    const long row0 = ((long)blockIdx.x * WAVES + wid) * RPW;

    const unsigned buf0 = (unsigned)(uintptr_t)&xbuf[wid][0][0];
    const unsigned buf1 = (unsigned)(uintptr_t)&xbuf[wid][1][0];

    // Kick off first row DMA before the barrier (writes only xbuf).
    tdm_load_row(buf0, x + row0 * N_FEAT);

    __syncthreads();

    const float fw = fcw[0];
    const float fb = fcb[0];

    for (int it = 0; it < RPW; ++it) {
        if (it + 1 < RPW) {
            // Fence this wave's outstanding DS reads of the buffer we are
            // about to overwrite, then prefetch next row.
            asm volatile("s_wait_dscnt 0" ::: "memory");
            tdm_load_row(((it + 1) & 1) ? buf1 : buf0,
                         x + (row0 + it + 1) * N_FEAT);
            __builtin_amdgcn_s_wait_tensorcnt(1);  // row `it` complete (in-order)
        } else {
            __builtin_amdgcn_s_wait_tensorcnt(0);
        }
        asm volatile("" ::: "memory");

        const char*   cb    = (it & 1) ? (const char*)&xbuf[wid][1][0]
                                       : (const char*)&xbuf[wid][0][0];
        const float4* chunk = (const float4*)(cb + lane * CHUNK_STRIDE);

        float v[16];
        // Level 1: 32 -> 16 in-lane. Elems [32L,32L+32) -> pairs 16L..16L+15, off 0.
        #pragma unroll
        for (int j = 0; j < 8; ++j) {
            float4 q  = chunk[j];
            float4 c0 = lwq[16 * lane + 2 * j];
            float4 c1 = lwq[16 * lane + 2 * j + 1];
            v[2 * j]     = combine(q.x, q.y, c0.x, c0.y, c0.z, c0.w);
            v[2 * j + 1] = combine(q.z, q.w, c1.x, c1.y, c1.z, c1.w);
        }
        // Level 2: 16 -> 8, pairs 8L+j, off 512.
        #pragma unroll
        for (int j = 0; j < 8; ++j) {
            float4 c = lwq[512 + 8 * lane + j];
            v[j] = combine(v[2 * j], v[2 * j + 1], c.x, c.y, c.z, c.w);
        }
        // Level 3: 8 -> 4, pairs 4L+j, off 768.
        #pragma unroll
        for (int j = 0; j < 4; ++j) {
            float4 c = lwq[768 + 4 * lane + j];
            v[j] = combine(v[2 * j], v[2 * j + 1], c.x, c.y, c.z, c.w);
        }
        // Level 4: 4 -> 2, pairs 2L+j, off 896.
        #pragma unroll
        for (int j = 0; j < 2; ++j) {
            float4 c = lwq[896 + 2 * lane + j];
            v[j] = combine(v[2 * j], v[2 * j + 1], c.x, c.y, c.z, c.w);
        }
        // Level 5: 2 -> 1, pair L, off 960.
        float s;
        {
            float4 c = lwq[960 + lane];
            s = combine(v[0], v[1], c.x, c.y, c.z, c.w);
        }

        // Levels 6-10: cross-lane; lane i (i<m) combines lanes 2i,2i+1.
        int off = 992;
        #pragma unroll
        for (int m = 16; m >= 1; m >>= 1) {
            float x0 = __shfl(s, 2 * lane, 32);
            float x1 = __shfl(s, 2 * lane + 1, 32);
            int   p  = off + ((lane < m) ? lane : (m - 1));  // clamp for idle lanes
            float4 c = lwq[p];
            s = combine(x0, x1, c.x, c.y, c.z, c.w);
            off += m;
        }

        if (lane == 0) out[row0 + it] = sigmoidf_dev(s * fw + fb);
    }
}

extern "C" void kernel_launch(void* const* d_in, const int* in_sizes, int n_in,
                              void* d_out, int out_size, void* d_ws, size_t ws_size,
                              hipStream_t stream) {
    const float* x   = (const float*)d_in[0];
    const float* w   = (const float*)d_in[1];
    const float* b   = (const float*)d_in[2];
    const float* fcw = (const float*)d_in[3];
    const float* fcb = (const float*)d_in[4];
    float* out = (float*)d_out;

    // 65536 rows = 2048 blocks * 4 waves * 8 rows/wave
    tree_logic_kernel<<<2048, TPB, 0, stream>>>(x, w, b, fcw, fcb, out);
}